// IGNN_23390391894783
// MI455X (gfx1250) — compile-verified
//
#include <hip/hip_runtime.h>
#include <hip/hip_bf16.h>

typedef __attribute__((ext_vector_type(2))) float v2f;
typedef __attribute__((ext_vector_type(8))) float v8f;

#define KAPPA 0.9f
#define EPSF  1e-12f
#define KC    64   // K-chunk for B staging

// ---------------------------------------------------------------- utilities

__global__ void zero_kernel(float* p, long long count) {
  long long t = (long long)blockIdx.x * blockDim.x + threadIdx.x;
  if (t < count) p[t] = 0.0f;
}

__global__ void init_v_kernel(float* v, int n) {
  int t = blockIdx.x * blockDim.x + threadIdx.x;
  if (t < n) v[t] = 1.0f / sqrtf((float)n);
}

// (A v)[src] += w * v[dst]   (one thread per edge)
__global__ void matvec_kernel(float* av, const float* v, const float* w,
                              const int* src, const int* dst, int E) {
  int e = blockIdx.x * blockDim.x + threadIdx.x;
  if (e < E) unsafeAtomicAdd(&av[src[e]], w[e] * v[dst[e]]);
}

__global__ void sumsq_kernel(const float* x, int n, float* out) {
  __shared__ float red[256];
  int tid = threadIdx.x;
  int t = blockIdx.x * blockDim.x + tid;
  float s = 0.0f;
  for (int i = t; i < n; i += gridDim.x * blockDim.x) s += x[i] * x[i];
  red[tid] = s;
  __syncthreads();
  for (int o = 128; o > 0; o >>= 1) {
    if (tid < o) red[tid] += red[tid + o];
    __syncthreads();
  }
  if (tid == 0) unsafeAtomicAdd(out, red[0]);
}

__global__ void scale_v_kernel(float* v, const float* av, const float* S, int n) {
  int t = blockIdx.x * blockDim.x + threadIdx.x;
  if (t < n) v[t] = av[t] / (sqrtf(S[0]) + EPSF);
}

// out[c*rows + r] = in[r*cols + c]
__global__ void transpose_kernel(const float* in, float* out, int rows, int cols) {
  int t = blockIdx.x * blockDim.x + threadIdx.x;
  if (t < rows * cols) {
    int r = t / cols, c = t % cols;
    out[c * rows + r] = in[t];
  }
}

// ------------------------------------------- L1-ball row projection of W
__global__ void project_w_kernel(const float* W, const float* Sfin, float* Wt) {
  int m = blockIdx.x;      // row of W
  int k = threadIdx.x;     // 0..127
  float rho = sqrtf(Sfin[0]);
  float vrad = KAPPA / rho;
  float wv = W[m * 128 + k];

  __shared__ float sv[128];
  __shared__ float cs[128];
  __shared__ int cnt;
  if (k == 0) cnt = 0;
  sv[k] = fabsf(wv);
  __syncthreads();

  // bitonic sort, descending
  for (int size = 2; size <= 128; size <<= 1) {
    for (int stride = size >> 1; stride > 0; stride >>= 1) {
      __syncthreads();
      int ixj = k ^ stride;
      if (ixj > k) {
        float a = sv[k], b = sv[ixj];
        bool desc = ((k & size) == 0);
        if (desc ? (a < b) : (a > b)) { sv[k] = b; sv[ixj] = a; }
      }
    }
  }
  __syncthreads();

  // inclusive scan
  cs[k] = sv[k];
  __syncthreads();
  for (int off = 1; off < 128; off <<= 1) {
    float tv = (k >= off) ? cs[k - off] : 0.0f;
    __syncthreads();
    cs[k] += tv;
    __syncthreads();
  }

  if (sv[k] * (float)(k + 1) > cs[k] - vrad) atomicAdd(&cnt, 1);
  __syncthreads();

  int r = cnt;                                   // >= 1
  float theta = (cs[r - 1] - vrad) / (float)r;
  bool needs = cs[127] > vrad;
  float a = fabsf(wv);
  float sgn = (wv > 0.f) ? 1.f : ((wv < 0.f) ? -1.f : 0.f);
  float proj = sgn * fmaxf(a - theta, 0.f);
  Wt[k * 128 + m] = needs ? proj : wv;
}

// ------------------------------------------- SpMM edge scatter (node-major)
// Yt[dst][r] += w * Xt[src][r],  one thread per (edge, feature)
__global__ void spmm_kernel(const float* __restrict__ Xt, float* __restrict__ Yt,
                            const float* __restrict__ w, const int* __restrict__ src,
                            const int* __restrict__ dst, int E) {
  long long t = (long long)blockIdx.x * blockDim.x + threadIdx.x;
  long long e = t >> 7;
  int r = (int)(t & 127);
  if (e < (long long)E) {
    int s = src[e], d = dst[e];
    float val = w[e] * Xt[(size_t)s * 128 + r];
    unsafeAtomicAdd(&Yt[(size_t)d * 128 + r], val);
  }
}

// ------------------------------------------- WMMA f32 GEMM (LDS-resident)
// C(nRows x NOV) = A(nRows x K) * B(K x NOV) [+ bias] [relu]
// ACOL=false: A row-major, ld=K.  ACOL=true: A(j,k) at A[k*An + j].
// B row-major (K x NOV) in global. NO = NOV padded up to multiple of 16.
// Block: 256 threads = 8 waves; block = 16 rows x (NO) columns of C.
template <int K, int NO, int NOV, bool ACOL, bool BIAS, bool RELU>
__global__ __launch_bounds__(256) void wmma_gemm_kernel(
    const float* __restrict__ A, long long An,
    const float* __restrict__ B,
    const float* __restrict__ bias,
    float* __restrict__ C, int nRows) {
  constexpr int Kp  = K + 4;    // padded A stride (bank-conflict free, b64-aligned)
  constexpr int KCp = KC + 4;   // padded B chunk stride
  __shared__ float As[16 * Kp];      // As[m][k]  (A tile, transposed-to-rows)
  __shared__ float Bt[NO * KCp];     // Bt[col][k] for current K-chunk

  const int tid  = threadIdx.x;
  const int lane = tid & 31;
  const int wid  = tid >> 5;          // 0..7 -> column tile
  const int jt   = blockIdx.x * 16;   // row-tile base
  const int nt   = wid * 16;          // col-tile base
  const int m    = lane & 15;
  const int half = lane >> 4;

  // ---- stage A tile (16 x K) into LDS, coalesced
  if (ACOL) {
    #pragma unroll 4
    for (int idx = tid; idx < 16 * K; idx += 256) {
      int mm = idx & 15, kk = idx >> 4;
      int row = jt + mm; if (row > nRows - 1) row = nRows - 1;
      As[mm * Kp + kk] = A[(size_t)kk * An + row];
    }
  } else {
    #pragma unroll 4
    for (int idx = tid; idx < 16 * K; idx += 256) {
      int mm = idx / K, kk = idx % K;
      int row = jt + mm; if (row > nRows - 1) row = nRows - 1;
      As[mm * Kp + kk] = A[(size_t)row * K + kk];
    }
  }

  v8f acc = {};
  const float* ap = &As[m * Kp + 2 * half];

  for (int kc = 0; kc < K; kc += KC) {
    __syncthreads();
    // stage B chunk transposed: Bt[c][k] = B[(kc+k)*NOV + c]
    #pragma unroll 4
    for (int idx = tid; idx < NO * KC; idx += 256) {
      int c = idx % NO, kk = idx / NO;
      float bv = 0.0f;
      if (NO == NOV || c < NOV) bv = B[(size_t)(kc + kk) * NOV + c];
      Bt[c * KCp + kk] = bv;
    }
    __syncthreads();

    if (NO <= 128 && nt < NO) {   // wave-uniform
      const float* app = ap + kc;
      const float* bp  = &Bt[(nt + m) * KCp + 2 * half];
      #pragma unroll
      for (int k = 0; k < KC; k += 4) {
        v2f a = *(const v2f*)(app + k);   // ds_load_b64
        v2f b = *(const v2f*)(bp + k);    // ds_load_b64
        acc = __builtin_amdgcn_wmma_f32_16x16x4_f32(
            false, a, false, b, (short)0, acc, false, false);
      }
    }
  }

  if (nt < NO) {
    int col = nt + m;
    if (NO == NOV || col < NOV) {
      #pragma unroll
      for (int i = 0; i < 8; ++i) {
        int row = jt + i + 8 * half;
        if (row < nRows) {
          float val = acc[i];
          if (BIAS) val += bias[(size_t)row * NOV + col];
          if (RELU) val = fmaxf(val, 0.0f);
          C[(size_t)row * NOV + col] = val;
        }
      }
    }
  }
}

// ------------------------------------------- row L2-normalize (one wave/node)
__global__ void normalize_kernel(const float* X, float* H, int n) {
  int gw   = (blockIdx.x * blockDim.x + threadIdx.x) >> 5;
  int lane = threadIdx.x & 31;
  if (gw >= n) return;
  const float* row = X + (size_t)gw * 128;
  float x0 = row[lane], x1 = row[lane + 32], x2 = row[lane + 64], x3 = row[lane + 96];
  float s = x0 * x0 + x1 * x1 + x2 * x2 + x3 * x3;
  for (int o = 16; o > 0; o >>= 1) s += __shfl_xor(s, o, 32);
  float inv = 1.0f / fmaxf(sqrtf(s), EPSF);
  float* out = H + (size_t)gw * 128;
  out[lane] = x0 * inv; out[lane + 32] = x1 * inv;
  out[lane + 64] = x2 * inv; out[lane + 96] = x3 * inv;
}

// ---------------------------------------------------------------- launcher

extern "C" void kernel_launch(void* const* d_in, const int* in_sizes, int n_in,
                              void* d_out, int out_size, void* d_ws, size_t ws_size,
                              hipStream_t stream) {
  const float* feat = (const float*)d_in[0];   // (256, n) row-major
  const float* W    = (const float*)d_in[1];   // (128,128)
  const float* Om   = (const float*)d_in[2];   // (128,256)
  const float* Vw   = (const float*)d_in[3];   // (40,128)
  const float* ew   = (const float*)d_in[5];
  const int*   esrc = (const int*)d_in[6];
  const int*   edst = (const int*)d_in[7];

  const int n = in_sizes[4] / 128;
  const int E = in_sizes[5];
  const long long nm = (long long)n * 128;

  // ---- workspace carving (256B aligned)
  char* ws = (char*)d_ws;
  auto carve = [&](size_t bytes) {
    char* p = ws;
    ws += (bytes + 255) & ~(size_t)255;
    return p;
  };
  float* scal = (float*)carve(256);                 // [0] = sum-of-squares accum
  float* v    = (float*)carve((size_t)n * 4);
  float* av   = (float*)carve((size_t)n * 4);
  float* Wt   = (float*)carve(128 * 128 * 4);       // Wp^T
  float* Bo   = (float*)carve(256 * 128 * 4);       // Omega_1^T
  float* Bv   = (float*)carve(128 * 40 * 4);        // V_w^T
  float* bT   = (float*)carve((size_t)nm * 4);      // b^T (n x 128)
  float* Xa   = (float*)carve((size_t)nm * 4);
  float* Xb   = (float*)carve((size_t)nm * 4);
  float* Y    = (float*)carve((size_t)nm * 4);      // also U^T, then H

  const int TB = 256;
  dim3 nBlk((n + TB - 1) / TB);
  dim3 eBlk((E + TB - 1) / TB);
  dim3 nmBlk((unsigned)((nm + TB - 1) / TB));
  dim3 spmmBlk((unsigned)(((long long)E * 128 + TB - 1) / TB));
  dim3 rowTiles((n + 15) / 16);

  // ---- spectral radius: 50 power iterations + final matvec
  init_v_kernel<<<nBlk, TB, 0, stream>>>(v, n);
  for (int it = 0; it < 50; ++it) {
    zero_kernel<<<nBlk, TB, 0, stream>>>(av, n);
    matvec_kernel<<<eBlk, TB, 0, stream>>>(av, v, ew, esrc, edst, E);
    zero_kernel<<<1, 1, 0, stream>>>(scal, 1);
    sumsq_kernel<<<dim3(196), TB, 0, stream>>>(av, n, scal);
    scale_v_kernel<<<nBlk, TB, 0, stream>>>(v, av, scal, n);
  }
  zero_kernel<<<nBlk, TB, 0, stream>>>(av, n);
  matvec_kernel<<<eBlk, TB, 0, stream>>>(av, v, ew, esrc, edst, E);
  zero_kernel<<<1, 1, 0, stream>>>(scal, 1);
  sumsq_kernel<<<dim3(196), TB, 0, stream>>>(av, n, scal);   // scal[0]=rho^2

  // ---- project W rows onto L1 ball of radius kappa/rho; store Wp^T
  project_w_kernel<<<dim3(128), 128, 0, stream>>>(W, scal, Wt);

  // ---- transposed small operands
  transpose_kernel<<<dim3((128 * 256 + TB - 1) / TB), TB, 0, stream>>>(Om, Bo, 128, 256);
  transpose_kernel<<<dim3((40 * 128 + TB - 1) / TB), TB, 0, stream>>>(Vw, Bv, 40, 128);

  // ---- U^T (n x 128) = feat^T (n x 256) * Omega^T : A column-major, col stride n
  wmma_gemm_kernel<256, 128, 128, true, false, false>
      <<<rowTiles, 256, 0, stream>>>(feat, (long long)n, Bo, nullptr, Y, n);

  // ---- b^T = spmm(U)
  zero_kernel<<<nmBlk, TB, 0, stream>>>(bT, nm);
  spmm_kernel<<<spmmBlk, TB, 0, stream>>>(Y, bT, ew, esrc, edst, E);

  // ---- fixed-point iterations: X <- relu((XA)^T Wp^T + b^T)
  zero_kernel<<<nmBlk, TB, 0, stream>>>(Xa, nm);
  float* Xcur = Xa;
  float* Xnxt = Xb;
  for (int it = 0; it < 30; ++it) {
    zero_kernel<<<nmBlk, TB, 0, stream>>>(Y, nm);
    spmm_kernel<<<spmmBlk, TB, 0, stream>>>(Xcur, Y, ew, esrc, edst, E);
    wmma_gemm_kernel<128, 128, 128, false, true, true>
        <<<rowTiles, 256, 0, stream>>>(Y, 0LL, Wt, bT, Xnxt, n);
    float* t = Xcur; Xcur = Xnxt; Xnxt = t;
  }

  // ---- H = row-normalized X^T ; out = H * V_w^T
  normalize_kernel<<<dim3((n * 32 + TB - 1) / TB), TB, 0, stream>>>(Xcur, Y, n);
  wmma_gemm_kernel<128, 48, 40, false, false, false>
      <<<rowTiles, 256, 0, stream>>>(Y, 0LL, Bv, nullptr, (float*)d_out, n);
}